// Group_54941221650988
// MI455X (gfx1250) — compile-verified
//
#include <hip/hip_runtime.h>

typedef __attribute__((ext_vector_type(2))) float v2f;
typedef __attribute__((ext_vector_type(8))) float v8f;

#define kB 8
#define kN 16384
#define kG 512
#define kM 32

__device__ inline unsigned long long u64max(unsigned long long a, unsigned long long b) {
  return a > b ? a : b;
}

// Map float -> uint32 whose unsigned order equals float order (ascending).
__device__ inline unsigned orderKey(float f) {
  unsigned u = __float_as_uint(f);
  return (u & 0x80000000u) ? ~u : (u | 0x80000000u);
}

// ---------------- Kernel 1: farthest point sampling ----------------
// One workgroup per batch. 1024 threads, 16 points per thread held in VGPRs.
// dist update + packed-u64 argmax per iteration; 512 sequential iterations.
__global__ __launch_bounds__(1024) void fps_kernel(const float* __restrict__ xyz,
                                                   float* __restrict__ centers) {
  const int b = blockIdx.x;
  const float* X = xyz + (size_t)b * kN * 3;
  const int t = threadIdx.x;               // 0..1023
  const int lane = t & 31, wv = t >> 5;    // wave32

  float px[16], py[16], pz[16], dist[16];
#pragma unroll
  for (int i = 0; i < 16; ++i) {
    int p = t + (i << 10);
    px[i] = X[p * 3 + 0];
    py[i] = X[p * 3 + 1];
    pz[i] = X[p * 3 + 2];
    dist[i] = 1e10f;
  }

  __shared__ unsigned long long red[32];
  __shared__ float cent[3];

  int far = 0;  // reference's lax.scan emits previous 'farthest', starting at 0
  for (int it = 0; it < kG; ++it) {
    // owner of point 'far' publishes the centroid (static register indexing only)
#pragma unroll
    for (int i = 0; i < 16; ++i)
      if (far == t + (i << 10)) { cent[0] = px[i]; cent[1] = py[i]; cent[2] = pz[i]; }
    __syncthreads();
    if (t == 0) {
      centers[((size_t)b * kG + it) * 3 + 0] = cent[0];
      centers[((size_t)b * kG + it) * 3 + 1] = cent[1];
      centers[((size_t)b * kG + it) * 3 + 2] = cent[2];
    }
    const float cx = cent[0], cy = cent[1], cz = cent[2];

    unsigned long long best = 0ull;
#pragma unroll
    for (int i = 0; i < 16; ++i) {
      float dx = px[i] - cx, dy = py[i] - cy, dz = pz[i] - cz;
      float d = dx * dx + dy * dy + dz * dz;
      dist[i] = fminf(dist[i], d);
      // non-negative floats: bit pattern order == value order.
      // tie-break: ~idx so the SMALLEST index wins the max (matches jnp.argmax).
      unsigned long long key =
          ((unsigned long long)__float_as_uint(dist[i]) << 32) |
          (unsigned)(0xFFFFFFFFu - (unsigned)(t + (i << 10)));
      best = u64max(best, key);
    }
#pragma unroll
    for (int off = 16; off; off >>= 1)
      best = u64max(best, __shfl_xor(best, off, 32));
    if (lane == 0) red[wv] = best;
    __syncthreads();
    if (wv == 0) {
      unsigned long long v = red[lane];
#pragma unroll
      for (int off = 16; off; off >>= 1)
        v = u64max(v, __shfl_xor(v, off, 32));
      if (lane == 0) red[0] = v;
    }
    __syncthreads();
    far = (int)(0xFFFFFFFFu - (unsigned)(red[0] & 0xFFFFFFFFull));
    __syncthreads();
  }
}

// Branchless 16x16 score tile: every lane loads (x,y,z) of its column point,
// computes |p|^2 with FMAs, and cndmask-selects its B-fragment half. No exec
// divergence in the hot loop; one load clause per tile.
__device__ inline v8f score_tile(const float* __restrict__ X, int p0, int col,
                                 bool lowHalf, v2f afrag) {
  const float* P = X + (size_t)(p0 + col) * 3;
  float x = P[0], y = P[1], z = P[2];
  float n = __builtin_fmaf(x, x, __builtin_fmaf(y, y, z * z));
  v2f bfrag;
  bfrag[0] = lowHalf ? x : z;
  bfrag[1] = lowHalf ? y : n;
  v8f acc = {};
  return __builtin_amdgcn_wmma_f32_16x16x4_f32(false, afrag, false, bfrag,
                                               (short)0, acc, false, false);
}

// ---------------- Kernel 2: WMMA-scored radix-select top-32 + gather --------
// score(row=center, col=point) = |c|^2 - 2*c.p + |p|^2 (matches reference's
// square_distance up to rounding).  A(16x4) = (-2cx,-2cy,-2cz,1),
// B(4x16) = (px,py,pz,|p|^2) via V_WMMA_F32_16X16X4_F32, then + |c|^2 per row.
__global__ __launch_bounds__(512) void group_kernel(const float* __restrict__ xyz,
                                                    const float* __restrict__ color,
                                                    const float* __restrict__ centers,
                                                    float* __restrict__ neigh,
                                                    float* __restrict__ feats) {
  const int gt = blockIdx.x;     // 0..31, 16 groups each
  const int b  = blockIdx.y;     // 0..7
  const int t = threadIdx.x;     // 0..511 (16 waves)
  const int lane = t & 31, wv = t >> 5;
  const int g0 = gt * 16;
  const float* X = xyz   + (size_t)b * kN * 3;
  const float* C = color + (size_t)b * kN * 3;

  __shared__ unsigned int hist[16 * 256];
  __shared__ unsigned int prefixS[16];
  __shared__ int needS[16];
  __shared__ unsigned int kthS[16];
  __shared__ int clessS[16];
  __shared__ unsigned int cntA[16], cntB[16];
  __shared__ unsigned long long bufS[16 * kM];
  __shared__ float ccx[16], ccy[16], ccz[16], cc2[16];

  if (t < 16) {
    const float* cp = centers + ((size_t)b * kG + g0 + t) * 3;
    float x = cp[0], y = cp[1], z = cp[2];
    ccx[t] = x; ccy[t] = y; ccz[t] = z;
    cc2[t] = __builtin_fmaf(x, x, __builtin_fmaf(y, y, z * z));
    prefixS[t] = 0; needS[t] = kM;
  }
  for (int i = t; i < 16 * 256; i += 512) hist[i] = 0;
  __syncthreads();

  // A fragment (16x4 f32, 2 VGPRs): lanes 0-15 hold K=0,1; lanes 16-31 K=2,3.
  const bool lowHalf = lane < 16;
  v2f afrag;
  {
    int m = lane & 15;
    afrag[0] = lowHalf ? (-2.0f * ccx[m]) : (-2.0f * ccz[m]);
    afrag[1] = lowHalf ? (-2.0f * ccy[m]) : 1.0f;
  }

  const int col = lane & 15;
  const int rowBase = (lane >> 4) << 3;  // C layout: vgpr j = row rowBase+j

  // ---- four 8-bit radix passes to find exact 32nd-smallest key per row ----
  for (int pass = 0; pass < 4; ++pass) {
    const int shift = 24 - 8 * pass;
    const unsigned mask = pass ? (0xFFFFFFFFu << (32 - 8 * pass)) : 0u;

    for (int tt = wv; tt < kN / 16; tt += 16) {
      v8f acc = score_tile(X, tt * 16, col, lowHalf, afrag);
#pragma unroll
      for (int v = 0; v < 8; ++v) {
        int row = rowBase + v;
        unsigned key = orderKey(cc2[row] + acc[v]);
        if ((key & mask) == prefixS[row])
          atomicAdd(&hist[row * 256 + ((key >> shift) & 0xFF)], 1u);
      }
    }
    __syncthreads();
    if (lane == 0) {  // wave wv selects the digit for row wv
      int nd = needS[wv];
      unsigned cum = 0, d = 255;
      for (int bin = 0; bin < 256; ++bin) {
        unsigned c = hist[wv * 256 + bin];
        if (cum + c >= (unsigned)nd) { d = (unsigned)bin; break; }
        cum += c;
      }
      prefixS[wv] |= d << shift;
      needS[wv] = nd - (int)cum;
    }
    __syncthreads();
    for (int i = t; i < 16 * 256; i += 512) hist[i] = 0;
    __syncthreads();
  }

  if (lane == 0) {
    kthS[wv] = prefixS[wv];          // exact key of the 32nd-smallest element
    clessS[wv] = kM - needS[wv];     // # elements strictly below it
    cntA[wv] = 0; cntB[wv] = 0;
  }
  __syncthreads();

  // ---- collection pass (recompute scores via WMMA; L2-resident data) ----
  for (int tt = wv; tt < kN / 16; tt += 16) {
    const int p0 = tt * 16;
    v8f acc = score_tile(X, p0, col, lowHalf, afrag);
    const unsigned p = (unsigned)(p0 + col);
#pragma unroll
    for (int v = 0; v < 8; ++v) {
      int row = rowBase + v;
      unsigned key = orderKey(cc2[row] + acc[v]);
      unsigned kth = kthS[row];
      if (key < kth) {
        unsigned pos = atomicAdd(&cntA[row], 1u);
        if (pos < kM) bufS[row * kM + pos] = ((unsigned long long)key << 32) | p;
      } else if (key == kth) {
        unsigned pb = atomicAdd(&cntB[row], 1u);
        unsigned slot = (unsigned)clessS[row] + pb;
        if (slot < kM) bufS[row * kM + slot] = ((unsigned long long)key << 32) | p;
      }
    }
  }
  __syncthreads();

  // ---- per-row bitonic sort of 32 (key,idx) pairs: wave wv sorts row wv ----
  // Ascending (key, idx): matches lax.top_k order (ascending distance, stable).
  unsigned long long val = bufS[wv * kM + lane];
#pragma unroll
  for (int k = 2; k <= 32; k <<= 1) {
#pragma unroll
    for (int j = k >> 1; j > 0; j >>= 1) {
      unsigned long long o = __shfl_xor(val, j, 32);
      bool lower = ((lane & j) == 0);
      bool up    = ((lane & k) == 0);
      unsigned long long mn = val < o ? val : o;
      unsigned long long mx = val < o ? o : val;
      val = (lower == up) ? mn : mx;
    }
  }

  // ---- gather + write outputs ----
  {
    unsigned p = (unsigned)(val & 0xFFFFFFFFull);
    const float* P = X + (size_t)p * 3;
    const float* Q = C + (size_t)p * 3;
    float nx = P[0] - ccx[wv], ny = P[1] - ccy[wv], nz = P[2] - ccz[wv];
    float cr = Q[0], cg = Q[1], cb = Q[2];
    int g = g0 + wv;
    size_t e = ((size_t)b * kG + g) * kM + lane;
    neigh[e * 3 + 0] = nx; neigh[e * 3 + 1] = ny; neigh[e * 3 + 2] = nz;
    feats[e * 6 + 0] = nx; feats[e * 6 + 1] = ny; feats[e * 6 + 2] = nz;
    feats[e * 6 + 3] = cr; feats[e * 6 + 4] = cg; feats[e * 6 + 5] = cb;
  }
}

extern "C" void kernel_launch(void* const* d_in, const int* in_sizes, int n_in,
                              void* d_out, int out_size, void* d_ws, size_t ws_size,
                              hipStream_t stream) {
  const float* xyz   = (const float*)d_in[0];
  const float* color = (const float*)d_in[1];
  float* out = (float*)d_out;
  float* neigh   = out;                                  // B*G*M*3 = 393216
  float* centers = out + (size_t)kB * kG * kM * 3;       // B*G*3   = 12288
  float* feats   = centers + (size_t)kB * kG * 3;        // B*G*M*6 = 786432

  fps_kernel<<<kB, 1024, 0, stream>>>(xyz, centers);
  group_kernel<<<dim3(kG / 16, kB), 512, 0, stream>>>(xyz, color, centers, neigh, feats);

  (void)in_sizes; (void)n_in; (void)out_size; (void)d_ws; (void)ws_size;
}